// KirchhoffCurrentLaw_38010460570136
// MI455X (gfx1250) — compile-verified
//
#include <hip/hip_runtime.h>
#include <hip/hip_bf16.h>

typedef __attribute__((ext_vector_type(2))) float v2f;
typedef __attribute__((ext_vector_type(8))) float v8f;

#define KCL_EPS 1e-6f

// Kernel 1: zero the per-node accumulator (workspace) and the scalar output.
__global__ __launch_bounds__(256) void kcl_zero(float* __restrict__ ns,
                                                float* __restrict__ out, int n) {
    int i = blockIdx.x * 256 + threadIdx.x;
    if (i < n) ns[i] = 0.0f;
    if (i == 0) out[0] = 0.0f;
}

// Kernel 2: one thread per edge. Streamed edge data uses non-temporal loads
// (TH=NT) so the 179 MB edge stream doesn't evict the L2-resident node table
// (1.6 MB) and accumulator (400 KB). Gathers + f32 atomics stay in L2.
__global__ __launch_bounds__(256) void kcl_edges(const float* __restrict__ nf,
                                                 const long long* __restrict__ ei,
                                                 const float* __restrict__ ep,
                                                 const float* __restrict__ eprm,
                                                 float* __restrict__ ns, int E) {
    int e = blockIdx.x * 256 + threadIdx.x;
    if (e >= E) return;

    long long s64 = __builtin_nontemporal_load(&ei[(size_t)e]);
    long long d64 = __builtin_nontemporal_load(&ei[(size_t)E + (size_t)e]);
    int s = (int)s64;
    int d = (int)d64;

    // node_features row stride = 4 floats; we need the first two (re, im).
    const v2f* nf2 = (const v2f*)nf;
    v2f vs = nf2[(size_t)s * 2];
    v2f vd = nf2[(size_t)d * 2];

    float dx = vs.x - vd.x;
    float dy = vs.y - vd.y;

    float r = __builtin_nontemporal_load(&eprm[2 * (size_t)e]) + KCL_EPS;
    float x = __builtin_nontemporal_load(&eprm[2 * (size_t)e + 1]);
    float p = __builtin_nontemporal_load(&ep[(size_t)e]);

    // |dv| / |z| = sqrt(|dv|^2 / |z|^2): one sqrt instead of two.
    float num2 = dx * dx + dy * dy;
    float den2 = r * r + x * x;
    float I = __builtin_sqrtf(num2 / den2) * p;

    atomicAdd(&ns[d],  I);   // inflow at dst
    atomicAdd(&ns[s], -I);   // outflow at src
}

// Kernel 3: mean(node_sum^2). Per-lane partials in a grid-stride loop, then a
// wave32 reduction on the matrix pipe: D = A(16x4) x ones(4x16). Every column
// of D equals the sum of all 64 A slots, so with each lane's partial in one A
// slot, lane n and lane n+16 together hold the full sum across their 8 D VGPRs.
__global__ __launch_bounds__(256) void kcl_reduce(const float* __restrict__ ns,
                                                  float* __restrict__ out, int n) {
    float inv_n = 1.0f / (float)n;
    float acc = 0.0f;
    int stride = gridDim.x * 256;
    for (int i = blockIdx.x * 256 + threadIdx.x; i < n; i += stride) {
        float v = ns[i];
        acc = fmaf(v, v, acc);
    }

    v2f a; a.x = acc;  a.y = 0.0f;   // each lane's partial occupies one A slot
    v2f b; b.x = 1.0f; b.y = 1.0f;   // B = all-ones 4x16
    v8f c = {};
    // 8 args: (neg_a, A, neg_b, B, c_mod, C, reuse_a, reuse_b)
    v8f dm = __builtin_amdgcn_wmma_f32_16x16x4_f32(
        false, a, false, b, (short)0, c, false, false);

    float t = dm[0] + dm[1] + dm[2] + dm[3] + dm[4] + dm[5] + dm[6] + dm[7];

    unsigned lane = threadIdx.x & 31u;
    // lane n holds rows 0..7 of column n, lane n+16 holds rows 8..15:
    // the two halves of exactly one column sum -> add both, once per wave.
    if (lane == 0u || lane == 16u) {
        atomicAdd(out, t * inv_n);
    }
}

extern "C" void kernel_launch(void* const* d_in, const int* in_sizes, int n_in,
                              void* d_out, int out_size, void* d_ws, size_t ws_size,
                              hipStream_t stream) {
    const float*     nf   = (const float*)d_in[0];      // [N,4] f32
    const long long* ei   = (const long long*)d_in[1];  // [2,E] i64
    const float*     ep   = (const float*)d_in[2];      // [E]   f32
    const float*     eprm = (const float*)d_in[3];      // [E,2] f32
    float* out = (float*)d_out;
    float* ns  = (float*)d_ws;                          // [N] accumulator

    int N = in_sizes[0] / 4;   // node_features flat count / F
    int E = in_sizes[2];       // edge_probs count

    kcl_zero<<<(N + 255) / 256, 256, 0, stream>>>(ns, out, N);
    kcl_edges<<<(E + 255) / 256, 256, 0, stream>>>(nf, ei, ep, eprm, ns, E);
    kcl_reduce<<<128, 256, 0, stream>>>(ns, out, N);
}